// ModelNew_3556232922205
// MI455X (gfx1250) — compile-verified
//
#include <hip/hip_runtime.h>
#include <math.h>

typedef float v2f __attribute__((ext_vector_type(2)));
typedef float v8f __attribute__((ext_vector_type(8)));

// Problem constants (from reference): B=8, CIN=16, COUT=32, K=3, S=2, P=1, OP=1
// x: [8,16,32,32,32]  w: [16,32,3,3,3]  b,subtract: [32]  out: [8,32,32,32]
#define DI   32            // input spatial
#define PDIM 32            // pooled output spatial
#define XCIN_STRIDE 32768  // 32*32*32
#define WCIN_STRIDE 864    // 32*27

__global__ __launch_bounds__(256) void convt_pool_sm_swish_max(
    const float* __restrict__ x, const float* __restrict__ w,
    const float* __restrict__ bias, const float* __restrict__ sub,
    float* __restrict__ out)
{
    // per-strip convT results: [strip][pos][cout], padded to dodge bank conflicts
    __shared__ float sy[8][16][33];
    __shared__ float sz[16][33];

    const int tid  = threadIdx.x;
    const int lane = tid & 31;
    const int wv   = tid >> 5;        // strip id 0..7 (one per wave, wave32)
    const int pos  = lane & 15;       // N index (pooled-w position within tile)
    const int hi   = lane >> 4;       // lanes 16-31 hold K=2,3 / M=+8 halves

    int bid = blockIdx.x;
    const int pw0 = (bid & 1) << 4;  bid >>= 1;   // pooled-w tile base (0 or 16)
    const int ph  = bid & 31;        bid >>= 5;
    const int pd  = bid & 31;        bid >>= 5;
    const int b   = bid;                           // batch 0..7

    // strip parities: od = 2*pd+dp, oh = 2*ph+hp, ow = 2*pw+wp
    const int dp = (wv >> 2) & 1, hp = (wv >> 1) & 1, wp = wv & 1;

    // Tap lists (wave-uniform => uniform branches, EXEC stays all-ones for WMMA)
    int ndt = 0, kd_a[2], id_a[2];
    if (!dp) { kd_a[0] = 1; id_a[0] = pd; ndt = 1; }
    else {
        if (pd + 1 < DI) { kd_a[ndt] = 0; id_a[ndt] = pd + 1; ndt++; }
        kd_a[ndt] = 2; id_a[ndt] = pd; ndt++;
    }
    int nht = 0, kh_a[2], ih_a[2];
    if (!hp) { kh_a[0] = 1; ih_a[0] = ph; nht = 1; }
    else {
        if (ph + 1 < DI) { kh_a[nht] = 0; ih_a[nht] = ph + 1; nht++; }
        kh_a[nht] = 2; ih_a[nht] = ph; nht++;
    }
    int nwt = 0, kw_a[2], of_a[2];
    if (!wp) { kw_a[0] = 1; of_a[0] = 0; nwt = 1; }
    else     { kw_a[0] = 0; of_a[0] = 1; kw_a[1] = 2; of_a[1] = 0; nwt = 2; }

    // Per-lane boundary handling, hoisted: off=0 is always in-range; off=1 can
    // run one column past the input (iw==32 for pw==31). Use a CLAMPED address
    // (loads always execute -> no EXEC-mask branches) + v_cndmask zero-select.
    int   xcol_of[2];
    float msk_of[2];
    {
        const int iw0 = pw0 + pos;          // off = 0
        xcol_of[0] = iw0;      msk_of[0] = 1.0f;
        const int iw1 = iw0 + 1;            // off = 1
        const bool ok = (iw1 < DI);
        xcol_of[1] = ok ? iw1 : 0;
        msk_of[1]  = ok ? 1.0f : 0.0f;
    }

    v8f acc0 = {};   // couts 0..15  x positions 0..15
    v8f acc1 = {};   // couts 16..31 x positions 0..15

    const int cin_lo = hi ? 2 : 0;   // K-chunk sub-index for this half-wave

    for (int it = 0; it < ndt; ++it) {
        const int kd = kd_a[it], id = id_a[it];
        for (int jt = 0; jt < nht; ++jt) {
            const int kh = kh_a[jt], ih = ih_a[jt];
            for (int kt = 0; kt < nwt; ++kt) {
                const int   kw  = kw_a[kt];
                const int   off = of_a[kt];
                const float mk  = msk_of[off];
                const int   xbase = ((b * 16 /*CIN*/) * DI + id) * DI * DI
                                    + ih * DI + xcol_of[off];
                const int   wtap  = kd * 9 + kh * 3 + kw;
#pragma unroll
                for (int c = 0; c < 4; ++c) {
                    const int cin = 4 * c + cin_lo;
                    // B (4x16): lane<16 -> K=0(v0),1(v1); lane>=16 -> K=2,3
                    const int xi = xbase + cin * XCIN_STRIDE;
                    const float t0 = x[xi];
                    const float t1 = x[xi + XCIN_STRIDE];
                    v2f bv; bv.x = t0 * mk; bv.y = t1 * mk;
                    // A (16x4) = w[cin, cout, tap]: lane<16 -> K=0,1; lane>=16 -> K=2,3
                    const int wi0 = cin * WCIN_STRIDE + pos * 27 + wtap;   // couts 0..15
                    v2f a0; a0.x = w[wi0];  a0.y = w[wi0 + WCIN_STRIDE];
                    const int wi1 = wi0 + 16 * 27;                          // couts 16..31
                    v2f a1; a1.x = w[wi1];  a1.y = w[wi1 + WCIN_STRIDE];

                    acc0 = __builtin_amdgcn_wmma_f32_16x16x4_f32(
                        false, a0, false, bv, (short)0, acc0, false, false);
                    acc1 = __builtin_amdgcn_wmma_f32_16x16x4_f32(
                        false, a1, false, bv, (short)0, acc1, false, false);
                }
            }
        }
    }

    // C/D layout: VGPR i -> M = i (lanes 0-15) or i+8 (lanes 16-31), N = lane&15
    const int mofs = hi ? 8 : 0;
#pragma unroll
    for (int i = 0; i < 8; ++i) {
        sy[wv][pos][i + mofs]      = acc0[i];
        sy[wv][pos][16 + i + mofs] = acc1[i];
    }
    __syncthreads();

    // Max-pool across the 8 strips + bias - subtract  -> sz[pos][cout]
    for (int idx = tid; idx < 512; idx += 256) {
        const int p = idx >> 5, c = idx & 31;
        float v = sy[0][p][c];
#pragma unroll
        for (int s = 1; s < 8; ++s) v = fmaxf(v, sy[s][p][c]);
        sz[p][c] = v + bias[c] - sub[c];
    }
    __syncthreads();

    // Stable softmax over 32 channels with the reference's +1e-6*exp(-m) term,
    // then swish and channel-max. 16 positions -> threads 0..15.
    if (tid < 16) {
        const int p = tid;
        float zm = -INFINITY;
        for (int c = 0; c < 32; ++c) zm = fmaxf(zm, sz[p][c]);
        float s = 0.0f;
        for (int c = 0; c < 32; ++c) s += __expf(sz[p][c] - zm);
        const float denom = s + 1e-6f * __expf(-zm);
        float best = -INFINITY;
        for (int c = 0; c < 32; ++c) {
            const float pr = __expf(sz[p][c] - zm) / denom;
            const float sw = pr / (1.0f + __expf(-pr));   // sigmoid(pr)*pr
            best = fmaxf(best, sw);
        }
        out[((b * PDIM + pd) * PDIM + ph) * PDIM + pw0 + p] = best;
    }
}

extern "C" void kernel_launch(void* const* d_in, const int* in_sizes, int n_in,
                              void* d_out, int out_size, void* d_ws, size_t ws_size,
                              hipStream_t stream) {
    (void)in_sizes; (void)n_in; (void)d_ws; (void)ws_size; (void)out_size;
    const float* x    = (const float*)d_in[0];   // [8,16,32,32,32]
    const float* w    = (const float*)d_in[1];   // [16,32,3,3,3]
    const float* bias = (const float*)d_in[2];   // [32]
    const float* sub  = (const float*)d_in[3];   // [32]
    float* out = (float*)d_out;                  // [8,32,32,32]

    // blocks: 2 pw-tiles * 32 ph * 32 pd * 8 batches = 16384, 256 threads (8 waves)
    dim3 grid(2 * 32 * 32 * 8), block(256);
    hipLaunchKernelGGL(convt_pool_sm_swish_max, grid, block, 0, stream,
                       x, w, bias, sub, out);
}